// OptimizedAttention_83116207112889
// MI455X (gfx1250) — compile-verified
//
#include <hip/hip_runtime.h>
#include <math.h>

typedef __attribute__((ext_vector_type(2))) float v2f;
typedef __attribute__((ext_vector_type(8))) float v8f;
typedef __attribute__((ext_vector_type(4))) unsigned int u32x4;
typedef __attribute__((ext_vector_type(8))) int i32x8;
typedef __attribute__((ext_vector_type(4))) int i32x4;

#define WMMA_F32(a, b, c) \
    __builtin_amdgcn_wmma_f32_16x16x4_f32(false, (a), false, (b), (short)0, (c), false, false)

// ---------------------------------------------------------------------------
// TDM: 2-D tile load Global -> LDS (fp32 elements).
//   lds_off          : LDS byte offset of tile start
//   gptr             : global address of tile start
//   tile_w / tile_h  : tile dims (elements): w = contiguous dim0, h = rows
//   tensor_w/tensor_h: tensor dims (elements) for OOB
//   row_stride       : tensor dim0 stride (elements)
//   pad_en/pi/pa     : LDS padding (D# encodings)
// ---------------------------------------------------------------------------
__device__ __forceinline__ void tdm_load_2d(
    unsigned lds_off, const float* gptr,
    unsigned tile_w, unsigned tile_h,
    unsigned long long tensor_w, unsigned long long tensor_h,
    unsigned long long row_stride,
    unsigned pad_en, unsigned pi, unsigned pa)
{
    unsigned long long ga = (unsigned long long)gptr;
    u32x4 g0;
    g0.x = 1u;                                        // count=1, user desc
    g0.y = lds_off;                                   // LDS byte address
    g0.z = (unsigned)(ga & 0xffffffffu);              // global addr lo
    g0.w = (unsigned)((ga >> 32) & 0x01ffffffu) | (2u << 30);  // hi | type=2

    i32x8 g1;
    g1[0] = (int)((2u << 16) | (pad_en << 20) | (pi << 22) | (pa << 25)); // 4B elems
    g1[1] = (int)((tensor_w & 0xffffu) << 16);                         // dim0 lo16
    g1[2] = (int)(((tensor_w >> 16) & 0xffffu) | ((tensor_h & 0xffffu) << 16));
    g1[3] = (int)(((tensor_h >> 16) & 0xffffu) | ((tile_w & 0xffffu) << 16));
    g1[4] = (int)(tile_h & 0xffffu);                                   // tile_dim1, dim2=0
    g1[5] = (int)(row_stride & 0xffffffffu);                           // dim0_stride lo32
    g1[6] = (int)((row_stride >> 32) & 0xffffu);                       // hi16 | dim1_stride=0
    g1[7] = 0;

    i32x4 z4; z4[0] = 0; z4[1] = 0; z4[2] = 0; z4[3] = 0;
    i32x8 z8;
#pragma unroll
    for (int i = 0; i < 8; ++i) z8[i] = 0;
    __builtin_amdgcn_tensor_load_to_lds(g0, g1, z4, z4, z8, 0);
}

__device__ __forceinline__ unsigned lds_offset_of(const void* p) {
    // flat shared-aperture address: low 32 bits are the LDS byte offset
    return (unsigned)(unsigned long long)(uintptr_t)p;
}

// ---------------------------------------------------------------------------
// fp32 GEMM: C[M,N] = A[M,K] * B[K,N], row-major. Block = 256 threads = 8
// waves covering 128(M) x 64(N); K consumed in chunks of 32 (8 WMMA k-steps).
// A tile (128x32, padded rows of 34 dwords) and B tile (32x64) are moved by
// the Tensor Data Mover with double buffering; wave 0 drives the TDM.
// ---------------------------------------------------------------------------
#define KC 32
#define A_ROW 34               // 32 dwords + 2 pad (pad_interval=4, pad_amount=1)

__global__ __launch_bounds__(256) void gemm_f32_wmma(
    const float* __restrict__ A, const float* __restrict__ B,
    float* __restrict__ C, int M, int N, int K)
{
    __shared__ float ash[2][128 * A_ROW];
    __shared__ float bsh[2][KC * 64];

    const int tid  = threadIdx.x;
    const int wave = tid >> 5;
    const int lane = tid & 31;
    const int half = lane >> 4;
    const int l16  = lane & 15;

    const int mblk = blockIdx.y * 128;
    const int n0   = blockIdx.x * 64;
    const int S    = K / KC;

    v8f acc[4];
#pragma unroll
    for (int j = 0; j < 4; ++j)
#pragma unroll
        for (int r = 0; r < 8; ++r) acc[j][r] = 0.0f;

    const bool tdm_wave = (wave == 0);
    if (tdm_wave) {
        tdm_load_2d(lds_offset_of(&ash[0][0]), A + (size_t)mblk * K,
                    KC, 128, (unsigned long long)K, (unsigned long long)M,
                    (unsigned long long)K, 1u, 4u, 1u);
        tdm_load_2d(lds_offset_of(&bsh[0][0]), B + n0,
                    64, KC, (unsigned long long)N, (unsigned long long)K,
                    (unsigned long long)N, 0u, 0u, 0u);
    }

    const float* arow = &ash[0][0] + ((size_t)(wave * 16 + l16)) * A_ROW;

    for (int s = 0; s < S; ++s) {
        const int buf = s & 1;
        if (tdm_wave) {
            if (s + 1 < S) {
                const int k1 = (s + 1) * KC;
                tdm_load_2d(lds_offset_of(&ash[(s + 1) & 1][0]),
                            A + (size_t)mblk * K + k1,
                            KC, 128, (unsigned long long)K, (unsigned long long)M,
                            (unsigned long long)K, 1u, 4u, 1u);
                tdm_load_2d(lds_offset_of(&bsh[(s + 1) & 1][0]),
                            B + (size_t)k1 * N + n0,
                            64, KC, (unsigned long long)N, (unsigned long long)K,
                            (unsigned long long)N, 0u, 0u, 0u);
                __builtin_amdgcn_s_wait_tensorcnt(2);   // stage s has landed
            } else {
                __builtin_amdgcn_s_wait_tensorcnt(0);
            }
        }
        __syncthreads();

        const float* abuf = arow + (size_t)buf * (128 * A_ROW);
        const float* bbuf = &bsh[buf][0];
#pragma unroll
        for (int ss = 0; ss < KC / 4; ++ss) {
            // A fragment: lane = M row, half selects K pair
            v2f a = *(const v2f*)(abuf + ss * 4 + 2 * half);
#pragma unroll
            for (int j = 0; j < 4; ++j) {
                v2f b;
                b.x = bbuf[(ss * 4 + 2 * half) * 64 + j * 16 + l16];
                b.y = bbuf[(ss * 4 + 2 * half + 1) * 64 + j * 16 + l16];
                acc[j] = WMMA_F32(a, b, acc[j]);
            }
        }
        __syncthreads();
    }

    const int m0 = mblk + wave * 16;
#pragma unroll
    for (int j = 0; j < 4; ++j)
#pragma unroll
        for (int r = 0; r < 8; ++r)
            C[(size_t)(m0 + r + 8 * half) * N + n0 + j * 16 + l16] = acc[j][r];
}

// ---------------------------------------------------------------------------
// RoPE applied in place to q (heads 0..15) and k (kv heads 0..3) inside the
// packed qkv buffer [B, T, 3072]. rope: [T, 64, 2] = (cos, sin).
// ---------------------------------------------------------------------------
__global__ __launch_bounds__(256) void rope_kernel(
    float* __restrict__ qkv, const float* __restrict__ rope, int T)
{
    const int idx  = blockIdx.x * 256 + threadIdx.x;
    const int i    = idx & 63;
    const int head = (idx >> 6) % 20;
    const int t    = ((idx >> 6) / 20) % T;
    const int b    = idx / (64 * 20 * T);

    const size_t row = ((size_t)b * T + t) * 3072;
    const size_t off = row + (head < 16 ? head * 128 : 2048 + (head - 16) * 128) + 2 * i;

    const float c = rope[(t * 64 + i) * 2 + 0];
    const float s = rope[(t * 64 + i) * 2 + 1];
    const float x0 = qkv[off + 0];
    const float x1 = qkv[off + 1];
    qkv[off + 0] = x0 * c - x1 * s;
    qkv[off + 1] = x0 * s + x1 * c;
}

// ---------------------------------------------------------------------------
// Flash attention: one wave per (b, head, 16-query tile); 8 independent waves
// per block. Q in registers (A-layout, pre-scaled); S = Q*K^T (32 WMMAs),
// online softmax via 16-lane shfl reductions, P relayout D->A through padded
// LDS, O += P*V (32 WMMAs).
// ---------------------------------------------------------------------------
__global__ __launch_bounds__(256) void attn_kernel(
    const float* __restrict__ qkv, float* __restrict__ y, int T)
{
    __shared__ float pstage[8][16 * 17];

    const int tid  = threadIdx.x;
    const int wave = tid >> 5;
    const int lane = tid & 31;
    const int half = lane >> 4;
    const int l16  = lane & 15;

    const int QT   = T / 16;
    const int unit = blockIdx.x * 8 + wave;
    const int qt = unit % QT;
    const int h  = (unit / QT) % 16;
    const int b  = unit / (QT * 16);
    const int hk = h >> 2;                      // GQA: 4 q-heads per kv-head

    const size_t RS = 3072;
    const float scale = 0.08838834764831845f;   // 1/sqrt(128)

    const float* qbase = qkv + ((size_t)b * T + qt * 16) * RS + h * 128;
    v2f qa[32];
#pragma unroll
    for (int kk = 0; kk < 32; ++kk) {
        const float* p = qbase + (size_t)l16 * RS + kk * 4 + 2 * half;
        qa[kk].x = p[0] * scale;
        qa[kk].y = p[1] * scale;
    }

    v8f acc[8];
#pragma unroll
    for (int j = 0; j < 8; ++j)
#pragma unroll
        for (int r = 0; r < 8; ++r) acc[j][r] = 0.0f;

    float mrow[8], lrow[8];
#pragma unroll
    for (int r = 0; r < 8; ++r) { mrow[r] = -1e30f; lrow[r] = 0.0f; }

    float* lp = &pstage[wave][0];

    for (int kt = 0; kt <= qt; ++kt) {
        const float* kbase = qkv + ((size_t)b * T + kt * 16) * RS + 2048 + hk * 128;
        const float* vbase = qkv + ((size_t)b * T + kt * 16) * RS + 2560 + hk * 128;
        if (kt + 1 <= qt) {   // prefetch next K/V tiles into L2/L0
            __builtin_prefetch(kbase + 16 * RS + (size_t)l16 * RS, 0, 1);
            __builtin_prefetch(vbase + 16 * RS + (size_t)l16 * RS, 0, 1);
        }

        // ---- S = Q * K^T : B fragment = K transposed (lane = key row)
        v8f s;
#pragma unroll
        for (int r = 0; r < 8; ++r) s[r] = 0.0f;
#pragma unroll
        for (int kk = 0; kk < 32; ++kk) {
            const float* p = kbase + (size_t)l16 * RS + kk * 4 + 2 * half;
            v2f bf; bf.x = p[0]; bf.y = p[1];
            s = WMMA_F32(qa[kk], bf, s);
        }

        if (kt == qt) {       // causal mask on diagonal tile
#pragma unroll
            for (int r = 0; r < 8; ++r)
                if (l16 > r + 8 * half) s[r] = -1e30f;
        }

        // ---- online softmax (row r+8*half lives in VGPR r of this half)
#pragma unroll
        for (int r = 0; r < 8; ++r) {
            float rm = s[r];
#pragma unroll
            for (int off = 1; off < 16; off <<= 1)
                rm = fmaxf(rm, __shfl_xor(rm, off, 32));
            const float mnew = fmaxf(mrow[r], rm);
            const float sc   = __expf(mrow[r] - mnew);
            const float pv   = __expf(s[r] - mnew);
            float rs = pv;
#pragma unroll
            for (int off = 1; off < 16; off <<= 1)
                rs += __shfl_xor(rs, off, 32);
            lrow[r] = lrow[r] * sc + rs;
            mrow[r] = mnew;
#pragma unroll
            for (int j = 0; j < 8; ++j) acc[j][r] *= sc;
            lp[(r + 8 * half) * 17 + l16] = pv;   // D-layout -> LDS (padded)
        }

        asm volatile("s_wait_dscnt 0" ::: "memory");

        // ---- O += P * V : P in A-layout from LDS, V as B fragments
#pragma unroll
        for (int kk2 = 0; kk2 < 4; ++kk2) {
            v2f pa;
            pa.x = lp[l16 * 17 + kk2 * 4 + 2 * half];
            pa.y = lp[l16 * 17 + kk2 * 4 + 2 * half + 1];
#pragma unroll
            for (int j = 0; j < 8; ++j) {
                const float* vp = vbase + (size_t)(kk2 * 4 + 2 * half) * RS + j * 16 + l16;
                v2f vf; vf.x = vp[0]; vf.y = vp[RS];
                acc[j] = WMMA_F32(pa, vf, acc[j]);
            }
        }
    }

    float* ybase = y + ((size_t)b * T + qt * 16) * 2048 + h * 128;
#pragma unroll
    for (int j = 0; j < 8; ++j)
#pragma unroll
        for (int r = 0; r < 8; ++r)
            ybase[(size_t)(r + 8 * half) * 2048 + j * 16 + l16] = acc[j][r] / lrow[r];
}

// ---------------------------------------------------------------------------
extern "C" void kernel_launch(void* const* d_in, const int* in_sizes, int n_in,
                              void* d_out, int out_size, void* d_ws, size_t ws_size,
                              hipStream_t stream) {
    const float* x     = (const float*)d_in[0];   // [2, 2048, 2048]
    const float* rope  = (const float*)d_in[1];   // [2048, 64, 2]
    const float* w_qkv = (const float*)d_in[2];   // [2048, 3072]
    const float* w_o   = (const float*)d_in[3];   // [2048, 2048]
    float*       out   = (float*)d_out;           // [2, 2048, 2048]

    const int B = 2, T = 2048, C = 2048, QKV = 3072;
    const int M = B * T;                          // 4096

    float* qkv = (float*)d_ws;                    // M * 3072 floats
    float* y   = qkv + (size_t)M * QKV;           // M * 2048 floats

    // 1) qkv = x @ w_qkv
    gemm_f32_wmma<<<dim3(QKV / 64, M / 128), 256, 0, stream>>>(x, w_qkv, qkv, M, QKV, C);

    // 2) RoPE on q + k in place
    const int pairs = B * T * 20 * 64;
    rope_kernel<<<pairs / 256, 256, 0, stream>>>(qkv, rope, T);

    // 3) causal GQA flash attention -> y
    const int units = B * 16 * (T / 16);          // 4096 waves
    attn_kernel<<<units / 8, 256, 0, stream>>>(qkv, y, T);

    // 4) out = y @ w_o
    gemm_f32_wmma<<<dim3(C / 64, M / 128), 256, 0, stream>>>(y, w_o, out, M, C, C);
}